// BatchedAttentionLayer_68332929679617
// MI455X (gfx1250) — compile-verified
//
#include <hip/hip_runtime.h>
#include <hip/hip_bf16.h>
#include <math.h>

#define N_NODES  50000
#define N_EDGES  800000
#define IN_DIM   128
#define NUM_HEADS 8
#define OUT_DIM  16
#define HD       128   // NUM_HEADS*OUT_DIM

typedef __attribute__((ext_vector_type(2))) float v2f;
typedef __attribute__((ext_vector_type(8))) float v8f;

// ---------------------------------------------------------------------------
// Fused Q/K/V projection: C = h @ W + b for three weight sets at once.
// Grid: N_NODES/16 blocks (50000 is exactly divisible), 256 threads = 8 waves.
// Each block stages a 16x128 tile of h in LDS; wave w computes the 16-column
// tile [16w,16w+16) of Q, K and V via V_WMMA_F32_16X16X4_F32, sharing the
// A-fragment across the three accumulators.
// ---------------------------------------------------------------------------
__global__ void __launch_bounds__(256) qkv_gemm_kernel(
    const float* __restrict__ h,
    const float* __restrict__ Wq, const float* __restrict__ bq,
    const float* __restrict__ Wk, const float* __restrict__ bk,
    const float* __restrict__ Wv, const float* __restrict__ bv,
    float* __restrict__ Q, float* __restrict__ K, float* __restrict__ V) {
  // Row padded 128 -> 132 floats: bank(r*132+k) = (4r+k)%64 => conflict-free
  // across the 16 rows read by an A-fragment.
  __shared__ float hs[16 * 132];

  const int m0 = blockIdx.x * 16;

  // Cooperative load of 16x128 h tile as float4 (512 float4s / 256 threads).
  {
    const float4* src4 = (const float4*)(h + (size_t)m0 * IN_DIM);
    for (int j = threadIdx.x; j < 512; j += 256) {
      float4 v = src4[j];
      int row = j >> 5;            // 32 float4 per row
      int col = (j & 31) << 2;
      *(float4*)(&hs[row * 132 + col]) = v;
    }
  }
  __syncthreads();

  const int lane  = threadIdx.x & 31;
  const int wave  = threadIdx.x >> 5;      // 0..7 -> column tile
  const int n0    = wave * 16;
  const int laneN = lane & 15;
  const int kHalf = (lane >> 4) << 1;      // lanes 0-15: K+{0,1}; 16-31: K+{2,3}

  v8f accQ = {}; v8f accK = {}; v8f accV = {};

  for (int k0 = 0; k0 < IN_DIM; k0 += 4) {
    const int kk = k0 + kHalf;
    // A fragment (16x4 f32, ISA 7.12.2 layout) from LDS: ds_load_b64
    v2f a = *(const v2f*)(&hs[laneN * 132 + kk]);
    // B fragments (4x16 f32): lane -> column n0+laneN, VGPR0/1 -> K=kk,kk+1
    v2f bfq, bfk, bfv;
    bfq[0] = Wq[(size_t)kk * HD + n0 + laneN];
    bfq[1] = Wq[(size_t)(kk + 1) * HD + n0 + laneN];
    bfk[0] = Wk[(size_t)kk * HD + n0 + laneN];
    bfk[1] = Wk[(size_t)(kk + 1) * HD + n0 + laneN];
    bfv[0] = Wv[(size_t)kk * HD + n0 + laneN];
    bfv[1] = Wv[(size_t)(kk + 1) * HD + n0 + laneN];

    accQ = __builtin_amdgcn_wmma_f32_16x16x4_f32(false, a, false, bfq, (short)0, accQ, false, false);
    accK = __builtin_amdgcn_wmma_f32_16x16x4_f32(false, a, false, bfk, (short)0, accK, false, false);
    accV = __builtin_amdgcn_wmma_f32_16x16x4_f32(false, a, false, bfv, (short)0, accV, false, false);
  }

  // C/D layout: VGPR r -> row M=r (lanes 0-15) or M=r+8 (lanes 16-31), col=laneN.
  const float biasQ = bq[n0 + laneN];
  const float biasK = bk[n0 + laneN];
  const float biasV = bv[n0 + laneN];
  const int   rowBase = m0 + ((lane >> 4) << 3);
  const int   col = n0 + laneN;
#pragma unroll
  for (int r = 0; r < 8; ++r) {
    size_t idx = (size_t)(rowBase + r) * HD + col;
    Q[idx] = accQ[r] + biasQ;
    K[idx] = accK[r] + biasK;
    V[idx] = accV[r] + biasV;
  }
}

// ---------------------------------------------------------------------------
// Edge phase: one wave32 per edge. Lane L holds dims [4L,4L+4); lanes 4h..4h+3
// cover head h exactly, so a 2-step shfl_xor reduction yields per-head scores.
// Scatter with global_atomic_add_f32 into wV (=d_out) and z.
// ---------------------------------------------------------------------------
__global__ void __launch_bounds__(256) edge_kernel(
    const float* __restrict__ Q, const float* __restrict__ K,
    const float* __restrict__ V, const int* __restrict__ src,
    const int* __restrict__ dst, float* __restrict__ wV,
    float* __restrict__ z, int nEdges) {
  const int wave = threadIdx.x >> 5;
  const int lane = threadIdx.x & 31;
  const int e = blockIdx.x * 8 + wave;
  if (e >= nEdges) return;

  const int sN = src[e];
  const int dN = dst[e];

  const float4 kv = ((const float4*)(K + (size_t)sN * HD))[lane];
  const float4 qv = ((const float4*)(Q + (size_t)dN * HD))[lane];

  float p = kv.x * qv.x + kv.y * qv.y + kv.z * qv.z + kv.w * qv.w;
  // reduce the 4 lanes of each head group
  p += __shfl_xor(p, 1, 32);
  p += __shfl_xor(p, 2, 32);

  float score = p * 0.25f;                       // 1/sqrt(OUT_DIM)
  score = fminf(5.0f, fmaxf(-5.0f, score));
  const float sE = expf(score);

  const float4 vv = ((const float4*)(V + (size_t)sN * HD))[lane];
  float* o = wV + (size_t)dN * HD + lane * 4;
  atomicAdd(o + 0, vv.x * sE);
  atomicAdd(o + 1, vv.y * sE);
  atomicAdd(o + 2, vv.z * sE);
  atomicAdd(o + 3, vv.w * sE);
  if ((lane & 3) == 0) atomicAdd(z + (size_t)dN * NUM_HEADS + (lane >> 2), sE);
}

__global__ void zero_kernel(float* __restrict__ p, int n) {
  int i = blockIdx.x * blockDim.x + threadIdx.x;
  if (i < n) p[i] = 0.0f;
}

__global__ void finalize_kernel(float* __restrict__ out,
                                const float* __restrict__ z, int n) {
  int i = blockIdx.x * blockDim.x + threadIdx.x;
  if (i < n) {
    int node = i >> 7;           // /128
    int head = (i >> 4) & 7;     // /16 % 8
    out[i] = out[i] / (z[node * NUM_HEADS + head] + 1e-6f);
  }
}

extern "C" void kernel_launch(void* const* d_in, const int* in_sizes, int n_in,
                              void* d_out, int out_size, void* d_ws, size_t ws_size,
                              hipStream_t stream) {
  // setup_inputs order: h, e, src, dst, Wq, bq, Wk, bk, We, be, Wv, bv
  const float* h   = (const float*)d_in[0];
  // d_in[1] = e      : unused (E_proj is dead in the reference)
  const int*   src = (const int*)d_in[2];
  const int*   dst = (const int*)d_in[3];
  const float* Wq  = (const float*)d_in[4];
  const float* bq  = (const float*)d_in[5];
  const float* Wk  = (const float*)d_in[6];
  const float* bk  = (const float*)d_in[7];
  // d_in[8] = We, d_in[9] = be : unused
  const float* Wv  = (const float*)d_in[10];
  const float* bv  = (const float*)d_in[11];

  float* out = (float*)d_out;

  // Workspace layout: Q | K | V | z  (19.6M floats = 78.4 MB)
  float* Q = (float*)d_ws;
  float* K = Q + (size_t)N_NODES * HD;
  float* V = K + (size_t)N_NODES * HD;
  float* z = V + (size_t)N_NODES * HD;

  const int nOut = N_NODES * HD;          // 6,400,000
  const int nZ   = N_NODES * NUM_HEADS;   // 400,000

  // Re-zero accumulators every call (buffers are poisoned / not re-poisoned).
  zero_kernel<<<(nOut + 255) / 256, 256, 0, stream>>>(out, nOut);
  zero_kernel<<<(nZ + 255) / 256, 256, 0, stream>>>(z, nZ);

  qkv_gemm_kernel<<<N_NODES / 16, 256, 0, stream>>>(h, Wq, bq, Wk, bk, Wv, bv,
                                                    Q, K, V);

  edge_kernel<<<(N_EDGES + 7) / 8, 256, 0, stream>>>(Q, K, V, src, dst, out, z,
                                                     N_EDGES);

  finalize_kernel<<<(nOut + 255) / 256, 256, 0, stream>>>(out, z, nOut);
}